// SimpleNEAgent_74320114090502
// MI455X (gfx1250) — compile-verified
//
#include <hip/hip_runtime.h>

// SimpleNEAgent scan for MI455X (gfx1250, wave32).
//
// One wave == 16 batch rows, owned end-to-end (rows are fully independent).
// Per neuron step, the (16 x 64) gathered activation tile times the 64-weight
// vector is computed with 16 V_WMMA_F32_16X16X4_F32 ops (4 independent
// accumulator chains, depth 4) in full f32 precision. Weights are broadcast
// across the A-matrix's M dimension so every D column holds the per-row dot
// product -> each lane's acc[0] directly holds its own row's result.
//
// Layouts (wave32):
//   A 16x4 f32 (documented): V0 = {K0 | lanes>=16: K2}, V1 = {K1 | K3}
//   B  4x16 f32 (half-wave K split, matching the sparse-B tables):
//        V0 = {K0 over N=lanes 0..15 | K2}, V1 = {K1 | K3}
//   C/D 16x16 f32: lane l, vgpr v -> D[v + 8*(l>=16)][l%16]
//
// Scheduling: the per-step critical chain is (prev sigmoid store) -> 32
// batched LDS gathers -> WMMA chain -> v_exp_f32 -> store. Weight/index
// operands for step pos+1 are staged through LDS and pre-read into rotating
// registers at the end of step pos; global prefetch runs 2 steps ahead.

typedef float v2f __attribute__((ext_vector_type(2)));
typedef float v8f __attribute__((ext_vector_type(8)));

#define BATCH 512
#define INPS  1024
#define NEUR  512
#define KFAN  64
#define NOUT  16
#define ROWS  16   // rows per wave (== WMMA M)

__global__ __launch_bounds__(32) void ne_scan_wmma_kernel(
    const float* __restrict__ x,        // (512, 1024) row-major
    const int*   __restrict__ in_idxs,  // (512, 64)
    const float* __restrict__ weights,  // (512, 64)
    float*       __restrict__ out)      // (16, 512) row-major
{
    extern __shared__ float smem[];
    float* act = smem;                          // [1536][16] column-major activations
    float* wst = act + (INPS + NEUR) * ROWS;    // [2][64]  double-buffered weight row
    int*   ist = (int*)(wst + 2 * KFAN);        // [2][64]  double-buffered index row

    const int lane  = threadIdx.x;              // 0..31, block == one wave
    const int rbase = blockIdx.x * ROWS;        // first batch row of this wave
    const int h     = lane >> 4;                // half-wave: K offset selector
    const int n     = lane & 15;                // row (N column of B / D)

    // ---- init: x into columns [0,1024), zeros into [1024,1536) ----
    for (int t = lane; t < ROWS * INPS; t += 32) {
        int col = t & (INPS - 1);
        int r   = t >> 10;
        act[col * ROWS + r] = x[(size_t)(rbase + r) * INPS + col];
    }
    for (int t = lane; t < ROWS * NEUR; t += 32) {
        act[INPS * ROWS + t] = 0.0f;
    }

    // ---- prologue: stage row 0 and pre-read it; prefetch rows 1 and 2 ----
    v2f  aw[16];                                // weights operands for current step
    int2 jj[16];                                // gather indices for current step
    {
        float2 w0 = ((const float2*)weights)[lane];
        int2   j0 = ((const int2*)in_idxs)[lane];
        wst[2 * lane] = w0.x; wst[2 * lane + 1] = w0.y;
        ist[2 * lane] = j0.x; ist[2 * lane + 1] = j0.y;
        #pragma unroll
        for (int i = 0; i < 16; ++i) {
            const int k0 = 4 * i + 2 * h;
            float2 af = *(const float2*)(wst + k0);
            aw[i].x = af.x; aw[i].y = af.y;
            jj[i]   = *(const int2*)(ist + k0);
        }
    }
    float2 wA = ((const float2*)(weights + (size_t)1 * KFAN))[lane];   // row 1
    int2   jA = ((const int2*)(in_idxs + (size_t)1 * KFAN))[lane];
    float2 wB = ((const float2*)(weights + (size_t)2 * KFAN))[lane];   // row 2
    int2   jB = ((const int2*)(in_idxs + (size_t)2 * KFAN))[lane];

    // ---- serial scan over neurons ----
    for (int pos = 0; pos < NEUR; ++pos) {
        // Phase 1: batch-issue all 32 LDS gathers for this step.
        v2f b[16];
        #pragma unroll
        for (int i = 0; i < 16; ++i) {
            b[i].x = act[jj[i].x * ROWS + n];
            b[i].y = act[jj[i].y * ROWS + n];
        }

        // Phase 2: 16 WMMAs over K=64, 4 independent accumulator chains.
        v8f acc[4] = {};
        #pragma unroll
        for (int i = 0; i < 16; ++i) {
            acc[i & 3] = __builtin_amdgcn_wmma_f32_16x16x4_f32(
                false, aw[i], false, b[i], (short)0, acc[i & 3], false, false);
        }

        // Phase 3: sigmoid + store. Every lane holds dot[lane & 15]; both
        // half-waves store the same value to the same address (benign).
        float val = (acc[0][0] + acc[1][0]) + (acc[2][0] + acc[3][0]);
        float sg  = 1.0f / (1.0f + __expf(-val));    // v_exp_f32
        act[(INPS + pos) * ROWS + (lane & 15)] = sg;

        // Phase 4: stage row pos+1 (prefetched >=2 steps ago) and pre-read it
        // into aw/jj for the next iteration; refill the prefetch pipeline.
        if (pos + 1 < NEUR) {
            const int   s  = (pos + 1) & 1;
            float*      wc = wst + s * KFAN;
            int*        ic = ist + s * KFAN;
            wc[2 * lane] = wA.x; wc[2 * lane + 1] = wA.y;
            ic[2 * lane] = jA.x; ic[2 * lane + 1] = jA.y;
            #pragma unroll
            for (int i = 0; i < 16; ++i) {
                const int k0 = 4 * i + 2 * h;
                float2 af = *(const float2*)(wc + k0);
                aw[i].x = af.x; aw[i].y = af.y;
                jj[i]   = *(const int2*)(ic + k0);
            }
            wA = wB; jA = jB;
            if (pos + 3 < NEUR) {
                wB = ((const float2*)(weights + (size_t)(pos + 3) * KFAN))[lane];
                jB = ((const int2*)(in_idxs + (size_t)(pos + 3) * KFAN))[lane];
            }
        }
    }

    // ---- emit out[o][b] = activs[b][I + N - 16 + o], out is (16, 512) ----
    for (int t = lane; t < NOUT * ROWS; t += 32) {
        int o = t >> 4;
        int r = t & 15;
        out[(size_t)o * BATCH + rbase + r] =
            act[(INPS + NEUR - NOUT + o) * ROWS + r];
    }
}

extern "C" void kernel_launch(void* const* d_in, const int* in_sizes, int n_in,
                              void* d_out, int out_size, void* d_ws, size_t ws_size,
                              hipStream_t stream) {
    (void)in_sizes; (void)n_in; (void)d_ws; (void)ws_size; (void)out_size;
    const float* x       = (const float*)d_in[0];
    const int*   in_idxs = (const int*)d_in[1];
    const float* weights = (const float*)d_in[2];
    float*       out     = (float*)d_out;

    const size_t smem_bytes =
        (size_t)((INPS + NEUR) * ROWS + 2 * KFAN) * sizeof(float) +
        (size_t)(2 * KFAN) * sizeof(int);  // 99,328 B < 320 KB/WGP

    ne_scan_wmma_kernel<<<dim3(BATCH / ROWS), dim3(32), smem_bytes, stream>>>(
        x, in_idxs, weights, out);
}